// ContextPredictTrainer_69707319214469
// MI455X (gfx1250) — compile-verified
//
#include <hip/hip_runtime.h>

// GIN forward, MI455X (gfx1250, wave32).
// N=20000 nodes, E=320000 edges, H=64, L=3. Everything L2-resident (~30MB << 192MB).
// Bottleneck: segment-sum atomics (41M f32 atomics/layer). GEMMs done in fp32 WMMA.

#define NN 20000
#define NE 320000
#define H  64
#define H2 128
#define EA 9
#define BN_EPS 1e-5f
#define A_LD 132   // padded LDS row stride (floats): multiple of 4 (16B align), 132%64=4 -> bank spread

typedef float v2f __attribute__((ext_vector_type(2)));
typedef float v4f __attribute__((ext_vector_type(4)));
typedef float v8f __attribute__((ext_vector_type(8)));

// ---------------- kernel: h0 = emb0[x] ----------------
__global__ void k_embed(const int* __restrict__ x, const float* __restrict__ emb0,
                        float* __restrict__ h) {
  int i = blockIdx.x * blockDim.x + threadIdx.x;       // over N*H
  if (i >= NN * H) return;
  int n = i >> 6, c = i & 63;
  h[i] = emb0[x[n] * H + c];
}

// ---- kernel: agg init = self-loop contributions (left: h[n], right: We[sl]*slt+be) ----
__global__ void k_init_agg(const float* __restrict__ h, const float* __restrict__ We_l,
                           const float* __restrict__ be_l, const int* __restrict__ sli,
                           const int* __restrict__ slt, float* __restrict__ agg) {
  int i = blockIdx.x * blockDim.x + threadIdx.x;       // over N*2H
  if (i >= NN * H2) return;
  int n = i >> 7, c = i & 127;
  float v;
  if (c < H) {
    v = h[n * H + c];
  } else {
    int j = c - H;
    v = fmaf(We_l[sli[0] * H + j], (float)slt[0], be_l[j]);
  }
  agg[i] = v;
}

// ---- kernel: per-edge scatter: agg[src][0:64]+=h[dst], agg[src][64:128]+=ea@We+be ----
// One wave per edge; lane L owns columns L and L+32 of each half.
__global__ void k_edges(const int* __restrict__ ei, const float* __restrict__ ea,
                        const float* __restrict__ We_l, const float* __restrict__ be_l,
                        const float* __restrict__ h, float* __restrict__ agg) {
  const int lane   = threadIdx.x & 31;
  const int wave   = (blockIdx.x * blockDim.x + threadIdx.x) >> 5;
  const int nwaves = (gridDim.x * blockDim.x) >> 5;
  const int j0 = lane, j1 = lane + 32;
  for (int e = wave; e < NE; e += nwaves) {
    const int s = ei[e];          // segment (src)
    const int d = ei[NE + e];     // gather  (dst)
    float acc0 = be_l[j0], acc1 = be_l[j1];
#pragma unroll
    for (int k = 0; k < EA; ++k) {
      const float a = ea[e * EA + k];            // wave-uniform, L0-broadcast
      acc0 = fmaf(a, We_l[k * H + j0], acc0);    // We is 2.3KB -> WGP$-resident
      acc1 = fmaf(a, We_l[k * H + j1], acc1);
    }
    const float hv0 = h[(size_t)d * H + j0];     // coalesced 128B gather
    const float hv1 = h[(size_t)d * H + j1];
    float* ag = agg + (size_t)s * H2;
    atomicAdd(ag + j0, hv0);                     // global_atomic_add_f32 (no return)
    atomicAdd(ag + j1, hv1);
    atomicAdd(ag + H + j0, acc0);
    atomicAdd(ag + H + j1, acc1);
  }
}

// ---- kernel: node MLP via fp32 WMMA. 32 nodes/block, 256 threads = 8 waves. ----
// GEMM1: Z1(32x128)=agg@W1, +b1, BN, ReLU -> LDS.  GEMM2: Z2(32x64)=R@W2, +b2, (ReLU).
__global__ __launch_bounds__(256, 2)
void k_node(const float* __restrict__ agg,
            const float* __restrict__ W1, const float* __restrict__ b1,
            const float* __restrict__ gam, const float* __restrict__ bet,
            const float* __restrict__ mean, const float* __restrict__ var,
            const float* __restrict__ W2, const float* __restrict__ b2,
            float* __restrict__ hout, int relu_out) {
  __shared__ float ldsA[32 * A_LD];
  __shared__ float ldsR[32 * A_LD];
  const int tid  = threadIdx.x;
  const int lane = tid & 31;
  const int w    = tid >> 5;
  const int half = lane >> 4;    // 0: lanes 0-15, 1: lanes 16-31
  const int m    = lane & 15;
  const int nodeBase = blockIdx.x * 32;

  // Stage A tile (32x128 f32) into padded LDS: 16 floats/thread as 4x b128.
  {
    const int row = tid >> 3;
    const int col = (tid & 7) * 16;
    const float* g = agg + (size_t)(nodeBase + row) * H2 + col;
    float* p = ldsA + row * A_LD + col;
#pragma unroll
    for (int i = 0; i < 4; ++i)
      *(v4f*)(p + 4 * i) = *(const v4f*)(g + 4 * i);
  }
  __syncthreads();

  // GEMM1: wave w -> row tile rt = w&1 (16 nodes), col tiles ct0, ct0+1.
  const int rt   = w & 1;
  const int ct0  = (w >> 1) * 2;
  const int col0 = ct0 * 16 + m;
  const int col1 = col0 + 16;
  v8f c0 = {}, c1 = {};
  const float* arow = ldsA + (16 * rt + m) * A_LD;
#pragma unroll 4
  for (int kk = 0; kk < 32; ++kk) {
    const int kb = kk * 4 + 2 * half;             // A 16x4 f32 layout: v0=K0/K2, v1=K1/K3
    v2f a = *(const v2f*)(arow + kb);             // ds_load_b64, conflict-free (4r banks)
    v2f bA, bB;                                   // B 4x16: same K striping, N across lanes
    bA.x = W1[(kb + 0) * H2 + col0];
    bA.y = W1[(kb + 1) * H2 + col0];
    bB.x = W1[(kb + 0) * H2 + col1];
    bB.y = W1[(kb + 1) * H2 + col1];
    c0 = __builtin_amdgcn_wmma_f32_16x16x4_f32(false, a, false, bA, (short)0, c0, false, false);
    c1 = __builtin_amdgcn_wmma_f32_16x16x4_f32(false, a, false, bB, (short)0, c1, false, false);
  }
  // Epilogue 1: z = (c + b1 - mean)*gamma/sqrt(var+eps) + beta = fma(c, s, t); ReLU -> ldsR
  {
    const float s0 = gam[col0] * rsqrtf(var[col0] + BN_EPS);
    const float t0 = fmaf(b1[col0] - mean[col0], s0, bet[col0]);
    const float s1 = gam[col1] * rsqrtf(var[col1] + BN_EPS);
    const float t1 = fmaf(b1[col1] - mean[col1], s1, bet[col1]);
    float* r = ldsR + (16 * rt + 8 * half) * A_LD;  // C/D layout: VGPR v -> M = v + 8*half
#pragma unroll
    for (int v = 0; v < 8; ++v) {
      float z0 = fmaf(c0[v], s0, t0); z0 = z0 > 0.f ? z0 : 0.f;
      float z1 = fmaf(c1[v], s1, t1); z1 = z1 > 0.f ? z1 : 0.f;
      r[v * A_LD + col0] = z0;
      r[v * A_LD + col1] = z1;
    }
  }
  __syncthreads();

  // GEMM2: Z2(32x64) = R(32x128) @ W2(128x64): 2 row tiles x 4 col tiles = 8 waves.
  const int rt2  = w >> 2;
  const int ct2  = w & 3;
  const int col2 = ct2 * 16 + m;
  v8f c2 = {};
  const float* rrow = ldsR + (16 * rt2 + m) * A_LD;
#pragma unroll 4
  for (int kk = 0; kk < 32; ++kk) {
    const int kb = kk * 4 + 2 * half;
    v2f a = *(const v2f*)(rrow + kb);
    v2f b;
    b.x = W2[(kb + 0) * H + col2];
    b.y = W2[(kb + 1) * H + col2];
    c2 = __builtin_amdgcn_wmma_f32_16x16x4_f32(false, a, false, b, (short)0, c2, false, false);
  }
  {
    const float bias = b2[col2];
    const int rowg = nodeBase + 16 * rt2 + 8 * half;
#pragma unroll
    for (int v = 0; v < 8; ++v) {
      float z = c2[v] + bias;
      if (relu_out) z = z > 0.f ? z : 0.f;
      hout[(size_t)(rowg + v) * H + col2] = z;
    }
  }
}

extern "C" void kernel_launch(void* const* d_in, const int* in_sizes, int n_in,
                              void* d_out, int out_size, void* d_ws, size_t ws_size,
                              hipStream_t stream) {
  (void)in_sizes; (void)n_in; (void)out_size; (void)ws_size;
  const int*   x       = (const int*)d_in[0];
  const int*   ei      = (const int*)d_in[1];    // [2, E] row-major
  const float* ea      = (const float*)d_in[2];  // [E, 9]
  const float* emb0    = (const float*)d_in[3];  // [2, H]
  const float* We      = (const float*)d_in[4];  // [L, 9, H]
  const float* be      = (const float*)d_in[5];  // [L, H]
  const float* W1      = (const float*)d_in[6];  // [L, 2H, 2H]
  const float* b1      = (const float*)d_in[7];  // [L, 2H]
  const float* gam     = (const float*)d_in[8];
  const float* bet     = (const float*)d_in[9];
  const float* bn_mean = (const float*)d_in[10];
  const float* bn_var  = (const float*)d_in[11];
  const float* W2      = (const float*)d_in[12]; // [L, 2H, H]
  const float* b2      = (const float*)d_in[13]; // [L, H]
  const int*   sli     = (const int*)d_in[14];
  const int*   slt     = (const int*)d_in[15];
  float* out = (float*)d_out;

  float* hA  = (float*)d_ws;          // N*H
  float* hB  = hA + (size_t)NN * H;   // N*H
  float* agg = hB + (size_t)NN * H;   // N*2H

  k_embed<<<(NN * H + 255) / 256, 256, 0, stream>>>(x, emb0, hA);

  float* hin = hA;
  for (int l = 0; l < 3; ++l) {
    const float* We_l = We + (size_t)l * EA * H;
    const float* be_l = be + (size_t)l * H;
    k_init_agg<<<(NN * H2 + 255) / 256, 256, 0, stream>>>(hin, We_l, be_l, sli, slt, agg);
    k_edges<<<2048, 256, 0, stream>>>(ei, ea, We_l, be_l, hin, agg);
    float* hout = (l == 2) ? out : (hin == hA ? hB : hA);
    k_node<<<NN / 32, 256, 0, stream>>>(agg,
        W1 + (size_t)l * H2 * H2, b1 + (size_t)l * H2,
        gam + (size_t)l * H2, bet + (size_t)l * H2,
        bn_mean + (size_t)l * H2, bn_var + (size_t)l * H2,
        W2 + (size_t)l * H2 * H, b2 + (size_t)l * H,
        hout, (l < 2) ? 1 : 0);
    hin = hout;
  }
}